// ResidualVectorQuantizer_25520695673419
// MI455X (gfx1250) — compile-verified
//
#include <hip/hip_runtime.h>
#include <cstddef>

// ---------------------------------------------------------------------------
// Residual VQ (L=8 layers) for MI455X / gfx1250, wave32, WMMA f32 16x16x4.
//   B=16, D=512, S=1500, E=8, K=1024, L=8, TAU=1
// Outputs (flat): xq_sum[B,D,S] | x_idxs[B,L,S] | x_logits[B*S,L,K] | commit | cbl
// xq_sum = x - xr_final (identity).  commit == cbloss in forward.
// Weights stored lane-major (k-major) in ws so A/B fragment loads coalesce;
// Wi padded to 16 rows so no EXEC-divergent loads in the K=512 hot loop.
// Logits / xq_sum stores are non-temporal (786 MB streaming, > L2).
// ---------------------------------------------------------------------------

typedef float v2f __attribute__((ext_vector_type(2)));
typedef float v8f __attribute__((ext_vector_type(8)));

constexpr int Bn = 16, Dn = 512, Sn = 1500, En = 8, Kn = 1024, Ln = 8;
constexpr int Tn = Bn * Sn;            // 24000 tokens
constexpr int NT_TILES = Tn / 16;      // 1500 token tiles (exact)
constexpr float EPSN = 1e-12f;

// ---- d_out layout (floats) ----
constexpr size_t OUT_XQSUM  = 0;
constexpr size_t OUT_IDXS   = (size_t)Bn * Dn * Sn;                  // 12,288,000
constexpr size_t OUT_LOGITS = OUT_IDXS + (size_t)Bn * Ln * Sn;       // +192,000
constexpr size_t OUT_COMMIT = OUT_LOGITS + (size_t)Tn * Ln * Kn;     // +196,608,000
constexpr size_t OUT_CBL    = OUT_COMMIT + 1;

// ---- workspace layout (float units), ~51.6 MB total ----
constexpr size_t WS_WIT  = 0;                            // [L,D,16] k-major, padded
constexpr size_t WS_WOT  = WS_WIT + (size_t)Ln*Dn*16;    // [L,E,D]  k-major
constexpr size_t WS_CNT  = WS_WOT + (size_t)Ln*En*Dn;    // [L,E,K]  k-major
constexpr size_t WS_CN2  = WS_CNT + (size_t)Ln*En*Kn;    // [L,K]
constexpr size_t WS_XP   = WS_CN2 + (size_t)Ln*Kn;       // [T,E]
constexpr size_t WS_XN   = WS_XP  + (size_t)Tn*En;       // [T,E]
constexpr size_t WS_XN2  = WS_XN  + (size_t)Tn*En;       // [T]
constexpr size_t WS_IDX  = WS_XN2 + (size_t)Tn;          // [T] ints
constexpr size_t WS_PART = WS_IDX + (size_t)Tn;          // [1500]
constexpr size_t WS_XR   = WS_PART + (size_t)NT_TILES;   // [B,D,S]

#define WMMA_F32X4(a, b, c) \
  __builtin_amdgcn_wmma_f32_16x16x4_f32(false, (a), false, (b), (short)0, (c), false, false)

__device__ __forceinline__ v2f load2(const float* p) {   // 8B-aligned b64 load
  return *(const v2f*)p;
}

// ---------------------------------------------------------------------------
// 0) Precompute weight-norm weights (transposed/padded) + normalized codebook.
// ---------------------------------------------------------------------------
__global__ __launch_bounds__(256) void rvq_precompute(
    const float* __restrict__ iv, const float* __restrict__ ig,
    const float* __restrict__ ov, const float* __restrict__ og,
    const float* __restrict__ cbk,
    float* __restrict__ WiT, float* __restrict__ WoT,
    float* __restrict__ cnT, float* __restrict__ cn2) {
  int r = blockIdx.x * blockDim.x + threadIdx.x;
  if (r < Ln * 16) {                        // WiT columns (E padded to 16)
    int l = r >> 4, e = r & 15;
    float* wt = WiT + (size_t)l * Dn * 16;
    if (e < En) {
      const float* vr = iv + ((size_t)l * En + e) * Dn;
      float n2 = 0.f;
      for (int d = 0; d < Dn; ++d) n2 += vr[d] * vr[d];
      float sc = ig[l * En + e] / sqrtf(n2);     // torch weight_norm: g*v/||v||
      for (int d = 0; d < Dn; ++d) wt[d * 16 + e] = vr[d] * sc;
    } else {
      for (int d = 0; d < Dn; ++d) wt[d * 16 + e] = 0.0f;   // zero pad rows 8..15
    }
    return;
  }
  r -= Ln * 16;
  if (r < Ln * Dn) {                        // WoT columns: norm over E
    int l = r / Dn, d = r - l * Dn;
    const float* vr = ov + (size_t)r * En;
    float n2 = 0.f;
    #pragma unroll
    for (int e = 0; e < En; ++e) n2 += vr[e] * vr[e];
    float sc = og[r] / sqrtf(n2);
    float* wt = WoT + (size_t)l * En * Dn;
    #pragma unroll
    for (int e = 0; e < En; ++e) wt[e * Dn + d] = vr[e] * sc;
    return;
  }
  r -= Ln * Dn;
  if (r < Ln * Kn) {                        // normalized codebook, k-major
    int l = r / Kn, k = r - l * Kn;
    const float* vr = cbk + (size_t)r * En;
    float n2 = 0.f;
    #pragma unroll
    for (int e = 0; e < En; ++e) n2 += vr[e] * vr[e];
    float inv = 1.0f / fmaxf(sqrtf(n2), EPSN);
    float s2 = 0.f;
    float* ct = cnT + (size_t)l * En * Kn;
    #pragma unroll
    for (int e = 0; e < En; ++e) { float v = vr[e] * inv; ct[e * Kn + k] = v; s2 += v * v; }
    cn2[r] = s2;
  }
}

// ---------------------------------------------------------------------------
// 1) x_proj = Wi(ExD) @ xr(DxT) + ib.  One wave per 16-token tile, K=512.
// A from WiT (lane-major, padded -> no divergence), B from xr (stride-S rows).
// Epilogue (lanes 0-15 own a token column) emits x_proj, xn, ||xn||^2.
// ---------------------------------------------------------------------------
__global__ __launch_bounds__(256) void rvq_proj(
    const float* __restrict__ xr, const float* __restrict__ WiT,
    const float* __restrict__ ib,
    float* __restrict__ xp, float* __restrict__ xn, float* __restrict__ xn2) {
  int wave = (int)((blockIdx.x * blockDim.x + threadIdx.x) >> 5);
  if (wave >= NT_TILES) return;
  int l     = threadIdx.x & 31;
  int col   = l & 15;            // A row (E idx) and B col (token) stripe
  int khalf = (l >> 4) * 2;      // K sub-lane per ISA 16x4 layout
  int t     = wave * 16 + col;
  int bt    = t / Sn, s = t - bt * Sn;
  size_t xbase = (size_t)bt * Dn * Sn + (size_t)s;

  v8f c = {};
  #pragma unroll 4
  for (int k = 0; k < Dn; k += 4) {
    int kk = k + khalf;
    v2f a, b;
    a.x = WiT[kk * 16 + col];            // 64B-contiguous across lanes
    a.y = WiT[(kk + 1) * 16 + col];
    b.x = xr[xbase + (size_t)kk * Sn];   // 64B-contiguous across lanes
    b.y = xr[xbase + (size_t)(kk + 1) * Sn];
    c = WMMA_F32X4(a, b, c);
  }
  if (l < 16) {                  // D-frag rows 0..7 == e=0..7 of token t
    float v[8]; float nrm2 = 0.f;
    #pragma unroll
    for (int r = 0; r < 8; ++r) { v[r] = c[r] + ib[r]; nrm2 += v[r] * v[r]; }
    float inv = 1.0f / fmaxf(sqrtf(nrm2), EPSN);
    float n2 = 0.f;
    #pragma unroll
    for (int r = 0; r < 8; ++r) {
      float xnr = v[r] * inv;
      xp[(size_t)t * En + r] = v[r];
      xn[(size_t)t * En + r] = xnr;
      n2 += xnr * xnr;
    }
    xn2[t] = n2;
  }
}

// ---------------------------------------------------------------------------
// 2) Logits/argmax + loss partials. One wave per 16-token tile, 64 code tiles.
// logits = 2*dot - ||xn||^2 - ||cn||^2 (TAU=1).  Non-temporal logits stores.
// Argmax with first-index tie-break (jnp.argmax), shfl-xor butterfly per half.
// ---------------------------------------------------------------------------
__global__ __launch_bounds__(256) void rvq_dist(
    const float* __restrict__ xn, const float* __restrict__ xn2,
    const float* __restrict__ xp,
    const float* __restrict__ cnT, const float* __restrict__ cn2,
    const float* __restrict__ cb,          // raw codebook layer slice [K,E]
    float* __restrict__ out, int* __restrict__ idxw,
    float* __restrict__ partials, int layer, int first) {
  int wave = (int)((blockIdx.x * blockDim.x + threadIdx.x) >> 5);
  if (wave >= NT_TILES) return;
  int l = threadIdx.x & 31;
  int col = l & 15, khalf = (l >> 4) * 2, rowbase = (l >> 4) * 8;
  int t0 = wave * 16;

  float xn2v[8];
  #pragma unroll
  for (int r = 0; r < 8; ++r) xn2v[r] = xn2[t0 + rowbase + r];

  const float* xrow = xn + (size_t)(t0 + col) * En;
  v2f a0 = load2(xrow + khalf);          // adjacent K pair -> b64
  v2f a1 = load2(xrow + khalf + 4);

  float best[8]; int bidx[8];
  #pragma unroll
  for (int r = 0; r < 8; ++r) { best[r] = -3.4e38f; bidx[r] = 0; }

  float* lgout = out + OUT_LOGITS;
  for (int nt = 0; nt < Kn / 16; ++nt) {
    int n = nt * 16 + col;
    v2f b0, b1;                          // cnT lane-major: coalesced rows
    b0.x = cnT[khalf * Kn + n];
    b0.y = cnT[(khalf + 1) * Kn + n];
    b1.x = cnT[(khalf + 4) * Kn + n];
    b1.y = cnT[(khalf + 5) * Kn + n];
    v8f c = {};
    c = WMMA_F32X4(a0, b0, c);
    c = WMMA_F32X4(a1, b1, c);
    float c2 = cn2[n];
    #pragma unroll
    for (int r = 0; r < 8; ++r) {
      int t = t0 + rowbase + r;
      float lg = 2.0f * c[r] - xn2v[r] - c2;     // = -dist / TAU
      __builtin_nontemporal_store(lg, &lgout[((size_t)t * Ln + layer) * Kn + n]);
      if (lg > best[r]) { best[r] = lg; bidx[r] = n; }  // ascending n keeps first max
    }
  }
  // reduce over the 16 lanes of each row-half (xor masks stay within half)
  #pragma unroll
  for (int mask = 1; mask <= 8; mask <<= 1) {
    #pragma unroll
    for (int r = 0; r < 8; ++r) {
      float ov = __shfl_xor(best[r], mask, 32);
      int   oi = __shfl_xor(bidx[r], mask, 32);
      if (ov > best[r] || (ov == best[r] && oi < bidx[r])) { best[r] = ov; bidx[r] = oi; }
    }
  }
  if (col == 0) {                 // lanes 0 and 16 write rows 0-7 / 8-15
    #pragma unroll
    for (int r = 0; r < 8; ++r) {
      int t = t0 + rowbase + r;
      int bt = t / Sn, s = t - bt * Sn;
      out[OUT_IDXS + ((size_t)bt * Ln + layer) * Sn + s] = (float)bidx[r];
      idxw[t] = bidx[r];
    }
  }
  // loss partial: commit==cbloss forward; sum ||x_proj - cb[idx]||^2 over tile
  float acc = 0.f;
  if ((l & 15) < 8) {             // lanes 0-7 rows 0-7, lanes 16-23 rows 8-15
    int r = l & 7;
    int bi = 0;
    #pragma unroll
    for (int rr = 0; rr < 8; ++rr) if (rr == r) bi = bidx[rr];
    int t = t0 + rowbase + r;
    const float* xpr = xp + (size_t)t * En;
    const float* cbr = cb + (size_t)bi * En;
    #pragma unroll
    for (int e = 0; e < En; ++e) { float d = xpr[e] - cbr[e]; acc += d * d; }
  }
  #pragma unroll
  for (int mask = 1; mask <= 16; mask <<= 1) acc += __shfl_xor(acc, mask, 32);
  if (l == 0) { if (first) partials[wave] = acc; else partials[wave] += acc; }
}

// ---------------------------------------------------------------------------
// 3) xq_out = Wo(Dx8) @ cb[idx](8xT) + ob ; xr_out = xr_in - xq_out.
// One wave per (16 D-rows x 16 tokens) tile: 32*1500 = 48000 waves.
// ---------------------------------------------------------------------------
__global__ __launch_bounds__(256) void rvq_out(
    const float* __restrict__ WoT, const float* __restrict__ ob,
    const float* __restrict__ cb, const int* __restrict__ idxw,
    const float* __restrict__ xr_in, float* __restrict__ xr_out) {
  int wave = (int)((blockIdx.x * blockDim.x + threadIdx.x) >> 5);
  if (wave >= (Dn / 16) * NT_TILES) return;
  int dtile = wave & 31, ttile = wave >> 5;
  int l = threadIdx.x & 31;
  int col = l & 15, khalf = (l >> 4) * 2, rowbase = (l >> 4) * 8;
  int db = dtile * 16, t0 = ttile * 16;

  v2f a0, a1;                           // WoT lane-major: coalesced rows
  a0.x = WoT[khalf * Dn + db + col];
  a0.y = WoT[(khalf + 1) * Dn + db + col];
  a1.x = WoT[(khalf + 4) * Dn + db + col];
  a1.y = WoT[(khalf + 5) * Dn + db + col];
  int bi = idxw[t0 + col];
  const float* crow = cb + (size_t)bi * En;
  v2f b0 = load2(crow + khalf);
  v2f b1 = load2(crow + khalf + 4);
  v8f c = {};
  c = WMMA_F32X4(a0, b0, c);
  c = WMMA_F32X4(a1, b1, c);

  int t = t0 + col;
  int bt = t / Sn, s = t - bt * Sn;
  #pragma unroll
  for (int r = 0; r < 8; ++r) {
    int drow = db + rowbase + r;
    size_t addr = ((size_t)bt * Dn + drow) * Sn + (size_t)s;
    float val = c[r] + ob[drow];
    xr_out[addr] = xr_in[addr] - val;    // xr stays temporal: reused next layer (L2)
  }
}

// ---------------------------------------------------------------------------
// 4) xq_sum = x - xr_final (streamed) + serial deterministic loss reduce.
// ---------------------------------------------------------------------------
__global__ __launch_bounds__(256) void rvq_final(
    const float* __restrict__ x, const float* __restrict__ xr,
    const float* __restrict__ partials, float* __restrict__ out) {
  size_t i = (size_t)blockIdx.x * blockDim.x + threadIdx.x;
  constexpr size_t N = (size_t)Bn * Dn * Sn;
  if (i < N) __builtin_nontemporal_store(x[i] - xr[i], &out[OUT_XQSUM + i]);
  if (blockIdx.x == 0 && threadIdx.x == 0) {
    float ssum = 0.f;
    for (int w = 0; w < NT_TILES; ++w) ssum += partials[w];
    float cval = ssum / (float)((size_t)Bn * En * Sn);   // mean over (E,S), then B
    out[OUT_COMMIT] = cval;
    out[OUT_CBL]    = cval;   // identical forward value
  }
}

// ---------------------------------------------------------------------------
extern "C" void kernel_launch(void* const* d_in, const int* in_sizes, int n_in,
                              void* d_out, int out_size, void* d_ws, size_t ws_size,
                              hipStream_t stream) {
  (void)in_sizes; (void)n_in; (void)out_size; (void)ws_size;
  const float* x   = (const float*)d_in[0];   // [B,D,S]
  const float* iv  = (const float*)d_in[1];   // [L,E,D]
  const float* ig  = (const float*)d_in[2];   // [L,E]
  const float* ib  = (const float*)d_in[3];   // [L,E]
  const float* ov  = (const float*)d_in[4];   // [L,D,E]
  const float* og  = (const float*)d_in[5];   // [L,D]
  const float* ob  = (const float*)d_in[6];   // [L,D]
  const float* cbk = (const float*)d_in[7];   // [L,K,E]
  float* out = (float*)d_out;
  float* ws  = (float*)d_ws;

  float* WiT  = ws + WS_WIT;
  float* WoT  = ws + WS_WOT;
  float* cnT  = ws + WS_CNT;
  float* cn2  = ws + WS_CN2;
  float* xp   = ws + WS_XP;
  float* xn   = ws + WS_XN;
  float* xn2  = ws + WS_XN2;
  int*   idxw = (int*)(ws + WS_IDX);
  float* part = ws + WS_PART;
  float* xr   = ws + WS_XR;

  constexpr int PRE_ROWS = Ln * 16 + Ln * Dn + Ln * Kn;   // 12416
  rvq_precompute<<<(PRE_ROWS + 255) / 256, 256, 0, stream>>>(
      iv, ig, ov, og, cbk, WiT, WoT, cnT, cn2);

  constexpr int TOK_BLOCKS = (NT_TILES * 32 + 255) / 256;        // 188
  constexpr int OUT_BLOCKS = ((Dn / 16) * NT_TILES * 32) / 256;  // 6000

  for (int i = 0; i < Ln; ++i) {
    const float* xr_in = (i == 0) ? x : xr;
    rvq_proj<<<TOK_BLOCKS, 256, 0, stream>>>(
        xr_in, WiT + (size_t)i * Dn * 16, ib + (size_t)i * En, xp, xn, xn2);
    rvq_dist<<<TOK_BLOCKS, 256, 0, stream>>>(
        xn, xn2, xp, cnT + (size_t)i * En * Kn, cn2 + (size_t)i * Kn,
        cbk + (size_t)i * Kn * En, out, idxw, part, i, (i == 0) ? 1 : 0);
    rvq_out<<<OUT_BLOCKS, 256, 0, stream>>>(
        WoT + (size_t)i * En * Dn, ob + (size_t)i * Dn,
        cbk + (size_t)i * Kn * En, idxw, xr_in, xr);
  }
  rvq_final<<<((size_t)Bn * Dn * Sn + 255) / 256, 256, 0, stream>>>(x, xr, part, out);
}